// DCSAGE_GRU_73787538145687
// MI455X (gfx1250) — compile-verified
//
#include <hip/hip_runtime.h>
#include <math.h>

// ---------------------------------------------------------------------------
// DCSAGE + GraphNorm + GRU pipeline for MI455X (gfx1250, wave32)
//   - edge scatter/mean-aggregate : atomics (memory-bound phase)
//   - all GEMMs                   : V_WMMA_F32_16X16X4_F32 (full fp32 WMMA)
//   - weight staging              : Tensor Data Mover (tensor_load_to_lds)
//   - B-fragments                 : ds_load from addrspace(3) (not flat)
//   - GraphNorm                   : LDS block reductions + column atomics
// ---------------------------------------------------------------------------

typedef __attribute__((ext_vector_type(2))) float v2f;
typedef __attribute__((ext_vector_type(8))) float v8f;
typedef __attribute__((ext_vector_type(4))) unsigned int u32x4;
typedef __attribute__((ext_vector_type(8))) int i32x8;
typedef __attribute__((ext_vector_type(4))) int i32x4;

typedef __attribute__((address_space(3))) float lds_f;
typedef __attribute__((address_space(3))) const v2f lds_cv2f;

// --------------------------------- utils -----------------------------------

__global__ void zero_f32(float* __restrict__ p, long n) {
  long i = (long)blockIdx.x * blockDim.x + threadIdx.x;
  long stride = (long)gridDim.x * blockDim.x;
  for (; i < n; i += stride) p[i] = 0.0f;
}

// ------------------------- edge scatter (mean aggr) ------------------------

__global__ void scatter_kernel(const float* __restrict__ x,
                               const int* __restrict__ ei,   // [2, E]
                               const float* __restrict__ ea, // [E]
                               float* __restrict__ agg,      // [N, 64]
                               float* __restrict__ cnt,      // [N] or null
                               int E) {
  long idx = (long)blockIdx.x * blockDim.x + threadIdx.x;
  long total = (long)E * 16;
  if (idx >= total) return;
  int e = (int)(idx >> 4);
  int q = (int)(idx & 15);
  int src = ei[e];
  int dst = ei[E + e];
  float w = ea[e];
  const float* xs = x + (size_t)src * 64 + q * 4;
  __builtin_prefetch(xs, 0, 0);               // global_prefetch_b8
  float4 v = *(const float4*)xs;
  float* ap = agg + (size_t)dst * 64 + q * 4;
  atomicAdd(ap + 0, v.x * w);
  atomicAdd(ap + 1, v.y * w);
  atomicAdd(ap + 2, v.z * w);
  atomicAdd(ap + 3, v.w * w);
  if (cnt && q == 0) atomicAdd(cnt + dst, 1.0f);
}

// ---------------------- TDM 2D tile load: Global -> LDS --------------------
// D# per CDNA5 ISA: 64-row x K-col fp32 tile, row-major
// tensor with row stride ldw (elements), packed contiguously into LDS.
// clang-23 lane: 6-arg builtin (g0, g1, g2, g3, g4, cpol); groups 2..4 unused
// for a 2-D tensor.

__device__ inline void tdm_load_2d(unsigned lds_byte_addr, const void* gaddr,
                                   unsigned rows, unsigned cols,
                                   unsigned row_stride_elems) {
  unsigned long long ga = (unsigned long long)gaddr;
  u32x4 g0;
  g0.x = 1u;                                        // count=1 (valid user D#)
  g0.y = lds_byte_addr;                             // lds_addr  [63:32]
  g0.z = (unsigned)(ga & 0xFFFFFFFFu);              // global_addr lo
  g0.w = (unsigned)((ga >> 32) & 0x01FFFFFFu)       // global_addr[56:32]
       | (2u << 30);                                // type = 2 ("image")
  i32x8 g1;
  g1[0] = (int)(2u << 16);                          // data_size=4B; mask=0
  g1[1] = (int)((cols & 0xFFFFu) << 16);            // tensor_dim0[15:0]
  g1[2] = (int)(((cols >> 16) & 0xFFFFu)            // tensor_dim0[31:16]
       | ((rows & 0xFFFFu) << 16));                 // tensor_dim1[15:0]
  g1[3] = (int)(((rows >> 16) & 0xFFFFu)            // tensor_dim1[31:16]
       | ((cols & 0xFFFFu) << 16));                 // tile_dim0 = cols
  g1[4] = (int)(rows & 0xFFFFu);                    // tile_dim1 = rows, tile_dim2=0
  g1[5] = (int)row_stride_elems;                    // tensor_dim0_stride[31:0]
  g1[6] = 0;                                        // stride hi / dim1_stride lo
  g1[7] = 0;                                        // dim1_stride hi
  i32x4 z4 = {0, 0, 0, 0};                          // groups 2/3 unused (2-D)
  i32x8 z8 = {0, 0, 0, 0, 0, 0, 0, 0};              // group 4 unused
  __builtin_amdgcn_tensor_load_to_lds(g0, g1, z4, z4, z8, 0);
}

// ------------------------------ WMMA GEMM ----------------------------------
// Y[M x Nout] = X0*scale(rs0) @ W0[:, koff0:koff0+K0]^T
//             (+ X1 @ W1[:, koff1:koff1+K1]^T)  + bias
// grid = (M/16, Nout/64), block = 128 threads = 4 waves, 1 wave = 16x16 tile.
// Weight slabs DMA'd into LDS by the Tensor Data Mover; read back via ds_load.

__global__ __launch_bounds__(128)
void wmma_gemm(const float* __restrict__ X0, int ldx0, const float* __restrict__ rs0,
               const float* __restrict__ W0, int ldw0, int koff0, int K0,
               const float* __restrict__ X1, int ldx1,
               const float* __restrict__ W1, int ldw1, int koff1, int K1,
               const float* __restrict__ bias,
               float* __restrict__ Y, int ldy) {
  __shared__ float ldsW0[64 * 128];
  __shared__ float ldsW1[64 * 64];

  const int tid = threadIdx.x;
  const int nbase = blockIdx.y * 64;

  // wave 0 issues the TDM descriptors (EXEC-independent, one issue per wave),
  // waits on TENSORcnt, then the workgroup barrier publishes LDS to all waves.
  if (tid < 32) {
    tdm_load_2d((unsigned)(unsigned long long)(void*)ldsW0,
                W0 + (size_t)nbase * ldw0 + koff0, 64u, (unsigned)K0,
                (unsigned)ldw0);
    if (X1)
      tdm_load_2d((unsigned)(unsigned long long)(void*)ldsW1,
                  W1 + (size_t)nbase * ldw1 + koff1, 64u, (unsigned)K1,
                  (unsigned)ldw1);
    __builtin_amdgcn_s_wait_tensorcnt(0);
  }
  __syncthreads();

  // Launder 32-bit LDS byte offsets (TDM writes are invisible to alias
  // analysis) and rebuild addrspace(3) pointers so B-reads stay ds_load
  // with 32-bit addressing, not flat.
  unsigned w0off = (unsigned)(unsigned long long)(void*)ldsW0;
  unsigned w1off = (unsigned)(unsigned long long)(void*)ldsW1;
  asm volatile("" : "+v"(w0off), "+v"(w1off)::"memory");
  const lds_f* w0 = (const lds_f*)w0off;
  const lds_f* w1 = (const lds_f*)w1off;

  const int wv    = tid >> 5;        // wave id 0..3 -> 16-col slice
  const int lane  = tid & 31;
  const int lo    = lane & 15;
  const int hi    = lane >> 4;       // half-wave select
  const int klane = hi * 2;          // ISA 32-bit A layout: lanes16-31 hold K=2,3
  const long m    = (long)blockIdx.x * 16 + lo;   // A-frag row (node)
  const int  nl   = wv * 16 + lo;                 // B-frag row in LDS slab
  const int  ncol = nbase + nl;                   // output column of this lane

  v8f acc;
  const float bv = bias ? bias[ncol] : 0.0f;
#pragma unroll
  for (int r = 0; r < 8; ++r) acc[r] = bv;

  float sc = 1.0f;
  if (rs0) sc = 1.0f / fmaxf(rs0[m], 1.0f);       // mean-aggregation divide

  const float* xrow0 = X0 + (size_t)m * ldx0;
  for (int k = 0; k < K0; k += 4) {
    v2f a = *(const v2f*)(xrow0 + k + klane);
    a.x *= sc; a.y *= sc;
    v2f b = *(lds_cv2f*)(w0 + nl * K0 + k + klane);
    acc = __builtin_amdgcn_wmma_f32_16x16x4_f32(false, a, false, b,
                                                (short)0, acc, false, false);
  }
  if (X1) {
    const float* xrow1 = X1 + (size_t)m * ldx1;
    for (int k = 0; k < K1; k += 4) {
      v2f a = *(const v2f*)(xrow1 + k + klane);
      v2f b = *(lds_cv2f*)(w1 + nl * K1 + k + klane);
      acc = __builtin_amdgcn_wmma_f32_16x16x4_f32(false, a, false, b,
                                                  (short)0, acc, false, false);
    }
  }

  // D layout: VGPR r -> row (r + 8*hi), col lo  within the 16x16 tile
#pragma unroll
  for (int r = 0; r < 8; ++r) {
    long row = (long)blockIdx.x * 16 + r + 8 * hi;
    Y[row * ldy + ncol] = acc[r];
  }
}

// ------------------------------ GraphNorm ----------------------------------

__global__ void gn_colsum(const float* __restrict__ s, float* __restrict__ colsum, int M) {
  __shared__ float sm[256];
  const int tid = threadIdx.x;
  const int c = tid & 63;
  float acc = 0.0f;
  for (long r = (long)blockIdx.x * 4 + (tid >> 6); r < M; r += (long)gridDim.x * 4)
    acc += s[r * 64 + c];
  sm[tid] = acc;
  __syncthreads();
  if (tid < 64)
    atomicAdd(&colsum[c], sm[tid] + sm[tid + 64] + sm[tid + 128] + sm[tid + 192]);
}

__global__ void gn_colsq(const float* __restrict__ s, const float* __restrict__ colsum,
                         const float* __restrict__ ms, float* __restrict__ colsq, int M) {
  __shared__ float sm[256];
  const int tid = threadIdx.x;
  const int c = tid & 63;
  const float mean = colsum[c] / (float)M;
  const float shift = ms[c] * mean;
  float acc = 0.0f;
  for (long r = (long)blockIdx.x * 4 + (tid >> 6); r < M; r += (long)gridDim.x * 4) {
    float t = s[r * 64 + c] - shift;
    acc += t * t;
  }
  sm[tid] = acc;
  __syncthreads();
  if (tid < 64)
    atomicAdd(&colsq[c], sm[tid] + sm[tid + 64] + sm[tid + 128] + sm[tid + 192]);
}

__global__ void gn_apply(const float* __restrict__ s, const float* __restrict__ colsum,
                         const float* __restrict__ colsq, const float* __restrict__ ms,
                         const float* __restrict__ w, const float* __restrict__ b,
                         float* __restrict__ x_ws, float* __restrict__ xc,
                         int coloff, int M) {
  long i = (long)blockIdx.x * blockDim.x + threadIdx.x;
  if (i >= (long)M * 64) return;
  const int c = (int)(i & 63);
  const long r = i >> 6;
  const float mean = colsum[c] / (float)M;
  const float var  = colsq[c] / (float)M;
  float t = s[i] - ms[c] * mean;
  float y = w[c] * t * rsqrtf(var + 1e-5f) + b[c];
  y = fmaxf(y, 0.0f);
  x_ws[i] = y;
  xc[r * 128 + coloff + c] = y;
}

// ------------------------------- GRU gates ---------------------------------

__global__ void gru_elem(const float* __restrict__ gi, const float* __restrict__ gh,
                         const float* __restrict__ h, float* __restrict__ hn_out, int M) {
  long i = (long)blockIdx.x * blockDim.x + threadIdx.x;
  if (i >= (long)M * 64) return;
  const long n = i >> 6;
  const int j = (int)(i & 63);
  const float* gir = gi + n * 192;
  const float* ghr = gh + n * 192;
  float ir = gir[j], iz = gir[64 + j], inn = gir[128 + j];
  float hr = ghr[j], hz = ghr[64 + j], hnn = ghr[128 + j];
  float r = 1.0f / (1.0f + expf(-(ir + hr)));
  float z = 1.0f / (1.0f + expf(-(iz + hz)));
  float nn = tanhf(inn + r * hnn);
  hn_out[i] = (1.0f - z) * nn + z * h[i];
}

// ------------------------------- launcher ----------------------------------

extern "C" void kernel_launch(void* const* d_in, const int* in_sizes, int n_in,
                              void* d_out, int out_size, void* d_ws, size_t ws_size,
                              hipStream_t stream) {
  const float* x     = (const float*)d_in[0];
  const int*   ei    = (const int*)  d_in[1];
  const float* ea    = (const float*)d_in[2];
  const float* h1    = (const float*)d_in[3];
  const float* h2    = (const float*)d_in[4];
  const float* s1lW  = (const float*)d_in[5];
  const float* s1lb  = (const float*)d_in[6];
  const float* s1rW  = (const float*)d_in[7];
  const float* s2lW  = (const float*)d_in[8];
  const float* s2lb  = (const float*)d_in[9];
  const float* s2rW  = (const float*)d_in[10];
  const float* gn1w  = (const float*)d_in[11];
  const float* gn1b  = (const float*)d_in[12];
  const float* gn1ms = (const float*)d_in[13];
  const float* gn2w  = (const float*)d_in[14];
  const float* gn2b  = (const float*)d_in[15];
  const float* gn2ms = (const float*)d_in[16];
  const float* g1wih = (const float*)d_in[17];   // [192,128]
  const float* g1whh = (const float*)d_in[18];   // [192,64]
  const float* g1bih = (const float*)d_in[19];
  const float* g1bhh = (const float*)d_in[20];
  const float* g2wih = (const float*)d_in[21];   // [192,64]
  const float* g2whh = (const float*)d_in[22];
  const float* g2bih = (const float*)d_in[23];
  const float* g2bhh = (const float*)d_in[24];

  const int N = in_sizes[0] / 64;   // 100000 (divisible by 16)
  const int E = in_sizes[2];        // 1600000

  // workspace carve-up (floats)
  float* ws      = (float*)d_ws;
  float* aggbuf  = ws;  ws += (size_t)N * 64;
  float* sbuf    = ws;  ws += (size_t)N * 64;
  float* x1      = ws;  ws += (size_t)N * 64;
  float* x2      = ws;  ws += (size_t)N * 64;
  float* gi      = ws;  ws += (size_t)N * 192;
  float* gh      = ws;  ws += (size_t)N * 192;
  float* cnt     = ws;  ws += N;
  float* stats   = ws;  ws += 256;
  float* colsum1 = stats, *colsq1 = stats + 64;
  float* colsum2 = stats + 128, *colsq2 = stats + 192;

  float* out_xc = (float*)d_out;                       // [N,128]
  float* out_h1 = (float*)d_out + (size_t)N * 128;     // [N,64]
  float* out_h2 = (float*)d_out + (size_t)N * 192;     // [N,64]

  const unsigned gz   = (unsigned)(((size_t)N * 64 + 255) / 256);
  const unsigned gsc  = (unsigned)(((long)E * 16 + 255) / 256);
  dim3 g64(N / 16, 1), g192(N / 16, 3);

  // ---- SAGE layer 1 ----
  zero_f32<<<gz, 256, 0, stream>>>(aggbuf, (long)N * 64);
  zero_f32<<<(N + 255) / 256, 256, 0, stream>>>(cnt, N);
  zero_f32<<<1, 256, 0, stream>>>(stats, 256);
  scatter_kernel<<<gsc, 256, 0, stream>>>(x, ei, ea, aggbuf, cnt, E);
  wmma_gemm<<<g64, 128, 0, stream>>>(aggbuf, 64, cnt, s1lW, 64, 0, 64,
                                     x, 64, s1rW, 64, 0, 64, s1lb, sbuf, 64);
  gn_colsum<<<512, 256, 0, stream>>>(sbuf, colsum1, N);
  gn_colsq <<<512, 256, 0, stream>>>(sbuf, colsum1, gn1ms, colsq1, N);
  gn_apply <<<gz, 256, 0, stream>>>(sbuf, colsum1, colsq1, gn1ms, gn1w, gn1b,
                                    x1, out_xc, 0, N);

  // ---- SAGE layer 2 ----
  zero_f32<<<gz, 256, 0, stream>>>(aggbuf, (long)N * 64);
  scatter_kernel<<<gsc, 256, 0, stream>>>(x1, ei, ea, aggbuf, nullptr, E);
  wmma_gemm<<<g64, 128, 0, stream>>>(aggbuf, 64, cnt, s2lW, 64, 0, 64,
                                     x1, 64, s2rW, 64, 0, 64, s2lb, sbuf, 64);
  gn_colsum<<<512, 256, 0, stream>>>(sbuf, colsum2, N);
  gn_colsq <<<512, 256, 0, stream>>>(sbuf, colsum2, gn2ms, colsq2, N);
  gn_apply <<<gz, 256, 0, stream>>>(sbuf, colsum2, colsq2, gn2ms, gn2w, gn2b,
                                    x2, out_xc, 64, N);

  // ---- GRU cell 1 (input = concat(x1,x2), K=128 split into 2x64) ----
  wmma_gemm<<<g192, 128, 0, stream>>>(x1, 64, nullptr, g1wih, 128, 0, 64,
                                      x2, 64, g1wih, 128, 64, 64, g1bih, gi, 192);
  wmma_gemm<<<g192, 128, 0, stream>>>(h1, 64, nullptr, g1whh, 64, 0, 64,
                                      nullptr, 0, nullptr, 0, 0, 0, g1bhh, gh, 192);
  gru_elem<<<gz, 256, 0, stream>>>(gi, gh, h1, out_h1, N);

  // ---- GRU cell 2 ----
  wmma_gemm<<<g192, 128, 0, stream>>>(out_h1, 64, nullptr, g2wih, 64, 0, 64,
                                      nullptr, 0, nullptr, 0, 0, 0, g2bih, gi, 192);
  wmma_gemm<<<g192, 128, 0, stream>>>(h2, 64, nullptr, g2whh, 64, 0, 64,
                                      nullptr, 0, nullptr, 0, 0, 0, g2bhh, gh, 192);
  gru_elem<<<gz, 256, 0, stream>>>(gi, gh, h2, out_h2, N);
}